// CombinedLoss_78615081386299
// MI455X (gfx1250) — compile-verified
//
#include <hip/hip_runtime.h>
#include <hip/hip_bf16.h>

typedef float v2f __attribute__((ext_vector_type(2)));
typedef float v8f __attribute__((ext_vector_type(8)));

#define SMOOTH 0.1f
#define MARGIN 0.5f
#define PD_EPS 1e-6f

// ---------- helpers ----------
__device__ __forceinline__ unsigned int orderF(float f) {
    // monotone float -> uint mapping (total order)
    unsigned int u = __float_as_uint(f);
    return (u & 0x80000000u) ? ~u : (u | 0x80000000u);
}

__device__ __forceinline__ unsigned long long shflx64(unsigned long long v, int m) {
    unsigned int lo = (unsigned int)v;
    unsigned int hi = (unsigned int)(v >> 32);
    lo = __shfl_xor(lo, m, 32);
    hi = __shfl_xor(hi, m, 32);
    return ((unsigned long long)hi << 32) | lo;
}

// ---------- 0: init packed per-row accumulators ----------
__global__ void init_kernel(unsigned long long* posKey, unsigned long long* negKey, int B) {
    int i = blockIdx.x * blockDim.x + threadIdx.x;
    if (i < B) {
        posKey[i] = 0ull;                      // smaller than any real (val,idx) key
        negKey[i] = ~0ull;                     // larger than any real (val,idx) key
    }
}

// ---------- 1: row squared norms (one wave per row) ----------
__global__ void sq_kernel(const float* __restrict__ emb, float* __restrict__ sq, int D, int B) {
    int lane = threadIdx.x & 31;
    int row  = blockIdx.x * (blockDim.x >> 5) + (threadIdx.x >> 5);
    if (row >= B) return;
    const float* e = emb + (size_t)row * D;
    float s = 0.f;
    for (int k = lane; k < D; k += 32) { float v = e[k]; s += v * v; }
    for (int m = 16; m; m >>= 1) s += __shfl_xor(s, m, 32);
    if (lane == 0) sq[row] = s;
}

// ---------- 2: smoothed cross-entropy per row (block per row) ----------
__global__ void __launch_bounds__(256) ce_kernel(const float* __restrict__ logits,
                                                 const int* __restrict__ labels,
                                                 float* __restrict__ rowCE, int C) {
    int row = blockIdx.x;
    int tid = threadIdx.x;
    const float* x = logits + (size_t)row * C;
    __shared__ float r1[256];
    __shared__ float r2[256];

    float m = -3.402823466e+38f;
    for (int c = tid; c < C; c += 256) m = fmaxf(m, x[c]);
    r1[tid] = m; __syncthreads();
    for (int s = 128; s > 0; s >>= 1) {
        if (tid < s) r1[tid] = fmaxf(r1[tid], r1[tid + s]);
        __syncthreads();
    }
    m = r1[0]; __syncthreads();

    float se = 0.f, sx = 0.f;
    for (int c = tid; c < C; c += 256) { float v = x[c]; se += expf(v - m); sx += v; }
    r1[tid] = se; r2[tid] = sx; __syncthreads();
    for (int s = 128; s > 0; s >>= 1) {
        if (tid < s) { r1[tid] += r1[tid + s]; r2[tid] += r2[tid + s]; }
        __syncthreads();
    }
    if (tid == 0) {
        float lse    = m + logf(r1[0]);
        float xl     = x[labels[row]];
        float nll    = lse - xl;                 // -logp[label]
        float smooth = lse - r2[0] / (float)C;   // mean(-logp)
        rowCE[row]   = (1.0f - SMOOTH) * nll + SMOOTH * smooth;
    }
}

// ---------- 3: fused Gram (fp32 WMMA) + batch-hard mining ----------
// One wave computes a 16x64 tile of E·E^T via v_wmma_f32_16x16x4_f32 (A-frag
// reused over 4 B-frags), converts to d2 in registers, masks by labels/self,
// reduces argmax(pos)/argmin(neg) per row, and commits one packed 64-bit
// atomic per row per tile. The 256 MB distance matrix is never materialized.
__global__ void __launch_bounds__(256) gram_mine_kernel(
    const float* __restrict__ emb, const float* __restrict__ sq,
    const int* __restrict__ labels,
    unsigned long long* __restrict__ posKey, unsigned long long* __restrict__ negKey,
    int B, int D)
{
    const int lane = threadIdx.x & 31;
    const int wave = threadIdx.x >> 5;
    const int numColGroups = B >> 6;                 // 64-column groups
    const int task     = blockIdx.x * 8 + wave;
    const int rowTile  = task / numColGroups;
    const int colGroup = task % numColGroups;
    const int rowBase  = rowTile << 4;
    const int colBase  = colGroup << 6;

    const int half = lane >> 4;                      // 0: K 0..1 / rows 0..7, 1: K 2..3 / rows 8..15
    const int l15  = lane & 15;
    const int kOff = half * 2;

    const float* aPtr  = emb + (size_t)(rowBase +      l15) * D + kOff;
    const float* bPtr0 = emb + (size_t)(colBase +  0 + l15) * D + kOff;
    const float* bPtr1 = emb + (size_t)(colBase + 16 + l15) * D + kOff;
    const float* bPtr2 = emb + (size_t)(colBase + 32 + l15) * D + kOff;
    const float* bPtr3 = emb + (size_t)(colBase + 48 + l15) * D + kOff;

    v8f acc[4];
    acc[0] = v8f{}; acc[1] = v8f{}; acc[2] = v8f{}; acc[3] = v8f{};

    for (int k0 = 0; k0 < D; k0 += 4) {
        v2f a  = *(const v2f*)(aPtr  + k0);
        v2f b0 = *(const v2f*)(bPtr0 + k0);
        v2f b1 = *(const v2f*)(bPtr1 + k0);
        v2f b2 = *(const v2f*)(bPtr2 + k0);
        v2f b3 = *(const v2f*)(bPtr3 + k0);
        acc[0] = __builtin_amdgcn_wmma_f32_16x16x4_f32(false, a, false, b0, (short)0, acc[0], false, false);
        acc[1] = __builtin_amdgcn_wmma_f32_16x16x4_f32(false, a, false, b1, (short)0, acc[1], false, false);
        acc[2] = __builtin_amdgcn_wmma_f32_16x16x4_f32(false, a, false, b2, (short)0, acc[2], false, false);
        acc[3] = __builtin_amdgcn_wmma_f32_16x16x4_f32(false, a, false, b3, (short)0, acc[3], false, false);
    }

    // metadata for the 8 rows this lane's half owns (M = 8*half + r)
    const int rBase = rowBase + half * 8;
    float sqRow[8]; int labRow[8];
    #pragma unroll
    for (int r = 0; r < 8; ++r) { sqRow[r] = sq[rBase + r]; labRow[r] = labels[rBase + r]; }

    unsigned long long posK[8], negK[8];
    #pragma unroll
    for (int r = 0; r < 8; ++r) { posK[r] = 0ull; negK[r] = ~0ull; }

    #pragma unroll
    for (int c = 0; c < 4; ++c) {
        const int   j   = colBase + c * 16 + l15;    // this lane's column (N = lane&15)
        const int   lj  = labels[j];
        const float sqj = sq[j];
        #pragma unroll
        for (int r = 0; r < 8; ++r) {
            const int i = rBase + r;
            if (j == i) continue;                    // exclude self
            const float d2 = sqRow[r] + sqj - 2.0f * acc[c][r];
            if (lj == labRow[r]) {
                // argmax over positives: ~j in low bits so ties pick smallest j
                unsigned long long key =
                    ((unsigned long long)orderF(d2) << 32) | (unsigned int)(~(unsigned int)j);
                if (key > posK[r]) posK[r] = key;
            } else {
                // argmin over negatives: j in low bits so ties pick smallest j
                unsigned long long key =
                    ((unsigned long long)orderF(d2) << 32) | (unsigned int)j;
                if (key < negK[r]) negK[r] = key;
            }
        }
    }

    // reduce the 16 columns held across the 16 lanes of each half
    #pragma unroll
    for (int m = 1; m <= 8; m <<= 1) {
        #pragma unroll
        for (int r = 0; r < 8; ++r) {
            unsigned long long o = shflx64(posK[r], m);
            if (o > posK[r]) posK[r] = o;
            o = shflx64(negK[r], m);
            if (o < negK[r]) negK[r] = o;
        }
    }

    if (l15 == 0) {                                  // lanes 0 and 16 commit
        #pragma unroll
        for (int r = 0; r < 8; ++r) {
            atomicMax(&posKey[rBase + r], posK[r]);
            atomicMin(&negKey[rBase + r], negK[r]);
        }
    }
}

// ---------- 4: per-row triplet loss from mined indices (wave per row) ----------
__global__ void triplet_kernel(const float* __restrict__ emb,
                               const unsigned long long* __restrict__ posKey,
                               const unsigned long long* __restrict__ negKey,
                               float* __restrict__ tri, float* __restrict__ validf,
                               int D, int B)
{
    int lane = threadIdx.x & 31;
    int row  = blockIdx.x * (blockDim.x >> 5) + (threadIdx.x >> 5);
    if (row >= B) return;

    unsigned long long pk = posKey[row];
    unsigned long long nk = negKey[row];
    bool valid = (pk != 0ull) && (nk != ~0ull);
    int jp = valid ? (int)(~(unsigned int)pk) : row;
    int jn = valid ? (int)( (unsigned int)nk) : row;

    const float* a = emb + (size_t)row * D;
    const float* p = emb + (size_t)jp  * D;
    const float* n = emb + (size_t)jn  * D;
    float sap = 0.f, san = 0.f;
    for (int k = lane; k < D; k += 32) {
        float av = a[k];
        float dp = av - p[k] + PD_EPS; sap += dp * dp;   // torch: eps inside the norm
        float dn = av - n[k] + PD_EPS; san += dn * dn;
    }
    for (int m = 16; m; m >>= 1) { sap += __shfl_xor(sap, m, 32); san += __shfl_xor(san, m, 32); }
    if (lane == 0) {
        float per = fmaxf(sqrtf(sap) - sqrtf(san) + MARGIN, 0.f);
        tri[row]    = valid ? per : 0.f;
        validf[row] = valid ? 1.f : 0.f;
    }
}

// ---------- 5: deterministic final scalar reduce ----------
__global__ void __launch_bounds__(1024) final_kernel(const float* __restrict__ rowCE,
                                                     const float* __restrict__ tri,
                                                     const float* __restrict__ validf,
                                                     float* __restrict__ out, int B)
{
    __shared__ float sCE[1024];
    __shared__ float sT[1024];
    __shared__ float sV[1024];
    int tid = threadIdx.x;
    float a = 0.f, b = 0.f, c = 0.f;
    for (int i = tid; i < B; i += 1024) { a += rowCE[i]; b += tri[i]; c += validf[i]; }
    sCE[tid] = a; sT[tid] = b; sV[tid] = c; __syncthreads();
    for (int s = 512; s > 0; s >>= 1) {
        if (tid < s) { sCE[tid] += sCE[tid + s]; sT[tid] += sT[tid + s]; sV[tid] += sV[tid + s]; }
        __syncthreads();
    }
    if (tid == 0) {
        float cls  = sCE[0] / (float)B;
        float trip = (sV[0] > 0.f) ? (sT[0] / sV[0]) : 0.f;
        out[0] = cls + trip;
    }
}

extern "C" void kernel_launch(void* const* d_in, const int* in_sizes, int n_in,
                              void* d_out, int out_size, void* d_ws, size_t ws_size,
                              hipStream_t stream) {
    const float* logits = (const float*)d_in[0];
    const float* emb    = (const float*)d_in[1];
    const int*   labels = (const int*)d_in[2];

    const int B = in_sizes[2];            // 8192
    const int C = in_sizes[0] / B;        // 1000
    const int D = in_sizes[1] / B;        // 128

    // workspace layout (B * 32 bytes total)
    unsigned long long* posKey = (unsigned long long*)d_ws;
    unsigned long long* negKey = posKey + B;
    float* sq     = (float*)(negKey + B);
    float* rowCE  = sq + B;
    float* tri    = rowCE + B;
    float* validf = tri + B;

    init_kernel<<<(B + 255) / 256, 256, 0, stream>>>(posKey, negKey, B);
    sq_kernel<<<B / 8, 256, 0, stream>>>(emb, sq, D, B);
    ce_kernel<<<B, 256, 0, stream>>>(logits, labels, rowCE, C);

    const int numRowTiles  = B / 16;
    const int numColGroups = B / 64;
    gram_mine_kernel<<<(numRowTiles * numColGroups) / 8, 256, 0, stream>>>(
        emb, sq, labels, posKey, negKey, B, D);

    triplet_kernel<<<B / 8, 256, 0, stream>>>(emb, posKey, negKey, tri, validf, D, B);
    final_kernel<<<1, 1024, 0, stream>>>(rowCE, tri, validf, (float*)d_out, B);
}